// MultiBlockStructuredScoreNet_57870389347053
// MI455X (gfx1250) — compile-verified
//
#include <hip/hip_runtime.h>

#define BATCH 8192
#define NBLK  9
#define NN    256
#define HID   32
#define PMAX  8
#define ZW    (NBLK * NN)   // 2304 floats per z row
#define MT    32            // batch rows per workgroup (16 per wave-subtile x2)

#define AS1 __attribute__((address_space(1)))   // global
#define AS3 __attribute__((address_space(3)))   // LDS

typedef __attribute__((ext_vector_type(2))) float v2f;
typedef __attribute__((ext_vector_type(8))) float v8f;

typedef const AS1 float* gfp;   // read-only global
typedef AS1 float*       gfpw;  // writable global
typedef const AS3 float* sfp;   // read-only LDS
typedef AS3 float*       sfpw;  // writable LDS

static __device__ __forceinline__ v8f zero8() {
  v8f z = {0.f, 0.f, 0.f, 0.f, 0.f, 0.f, 0.f, 0.f};
  return z;
}

// D = A(16x4) * B(4x16) + C, fp32 WMMA (wave32). 8-arg form:
// (neg_a, A, neg_b, B, c_mod, C, reuse_a, reuse_b)
static __device__ __forceinline__ v8f wmma_f32(v2f a, v2f b, v8f c) {
  return __builtin_amdgcn_wmma_f32_16x16x4_f32(false, a, false, b, (short)0, c,
                                               false, false);
}

// ---- A fragments: lane holds (M = m0 + lane%16, K = k0 + 2*(lane/16) + {0,1})
//      -> per-lane contiguous float pair.
static __device__ __forceinline__ v2f ldA_g(gfp base, int stride, int m0, int k0,
                                            int lrow, int lhi) {
  return *(const AS1 v2f*)(base + (size_t)(m0 + lrow) * stride + k0 + 2 * lhi);
}
static __device__ __forceinline__ v2f ldA_s(sfp base, int m0, int k0, int lrow,
                                            int lhi) {
  return *(const AS3 v2f*)(base + (m0 + lrow) * HID + k0 + 2 * lhi);  // ds_load_b64
}

// ---- B fragment from row-major M[K][N]: lane (K = k0+2*lhi+{0,1}, N = n0+lane%16)
static __device__ __forceinline__ v2f ldB_rm(gfp B, int stride, int k0, int n0,
                                             int lrow, int lhi) {
  gfp p = B + (size_t)(k0 + 2 * lhi) * stride + n0 + lrow;
  v2f r;
  r.x = p[0];
  r.y = p[stride];
  return r;
}

// ---- B fragment from TRANSPOSE of row-major Mt[N][K]: B[k][n] = Mt[n][k]
//      -> per-lane contiguous pair (b64 load).
static __device__ __forceinline__ v2f ldB_tr(gfp Mt, int stride, int k0, int n0,
                                             int lrow, int lhi) {
  return *(const AS1 v2f*)(Mt + (size_t)(n0 + lrow) * stride + k0 + 2 * lhi);
}

static __device__ __forceinline__ float sigf(float x) {
  return __builtin_amdgcn_rcpf(1.0f + __expf(-x));  // v_exp + v_rcp, no div chain
}
static __device__ __forceinline__ float siluf(float x) { return x * sigf(x); }
static __device__ __forceinline__ float dsiluf(float x) {
  float s = sigf(x);
  return s * (1.0f + x * (1.0f - s));
}

// Forward + backward through block-k MLP for this wave's 32 rows.
// Leaves v = dE/d(a1) (32 rows x 32) in vbuf (row-major, stride HID), so the
// caller can form mlp_grad = V @ W1^T with WMMA.
static __device__ void mlp_v(gfp z, gfp gW1, gfp gb1, gfp gW2, gfp gb2, gfp gW3,
                             int k, int R, int lrow, int lhi, sfpw vbuf) {
  gfp Zk = z + (size_t)k * NN;            // row stride ZW
  gfp W1 = gW1 + (size_t)k * NN * HID;    // [256][32]
  gfp W2 = gW2 + (size_t)k * HID * HID;   // [32][32]

  v8f a1[2][2];  // pre-activations, kept live for backward
  v8f t[2][2];

  // ---- a1 = Z_k @ W1 + b1 ; h1 = silu(a1) -> vbuf ----
#pragma unroll
  for (int nt = 0; nt < 2; ++nt) {
    int n0 = nt * 16;
    v8f c0 = zero8(), c1 = zero8();
#pragma unroll 4
    for (int kk = 0; kk < NN; kk += 4) {
      v2f b = ldB_rm(W1, HID, kk, n0, lrow, lhi);
      c0 = wmma_f32(ldA_g(Zk, ZW, R, kk, lrow, lhi), b, c0);
      c1 = wmma_f32(ldA_g(Zk, ZW, R + 16, kk, lrow, lhi), b, c1);
    }
    float bias = gb1[k * HID + n0 + lrow];
#pragma unroll
    for (int g = 0; g < 8; ++g) {
      c0[g] += bias;
      c1[g] += bias;
    }
    a1[nt][0] = c0;
    a1[nt][1] = c1;
#pragma unroll
    for (int g = 0; g < 8; ++g) {
      vbuf[(g + 8 * lhi) * HID + n0 + lrow] = siluf(c0[g]);
      vbuf[(16 + g + 8 * lhi) * HID + n0 + lrow] = siluf(c1[g]);
    }
  }

  // ---- a2 = H1 @ W2 + b2 ; u = gW3 * silu'(a2) ----
#pragma unroll
  for (int nt = 0; nt < 2; ++nt) {
    int n0 = nt * 16;
    v8f c0 = zero8(), c1 = zero8();
#pragma unroll
    for (int kk = 0; kk < HID; kk += 4) {
      v2f b = ldB_rm(W2, HID, kk, n0, lrow, lhi);
      c0 = wmma_f32(ldA_s(vbuf, 0, kk, lrow, lhi), b, c0);
      c1 = wmma_f32(ldA_s(vbuf, 16, kk, lrow, lhi), b, c1);
    }
    float bias = gb2[k * HID + n0 + lrow];
    float g3 = gW3[k * HID + n0 + lrow];
#pragma unroll
    for (int g = 0; g < 8; ++g) {
      c0[g] = g3 * dsiluf(c0[g] + bias);
      c1[g] = g3 * dsiluf(c1[g] + bias);
    }
    t[nt][0] = c0;
    t[nt][1] = c1;
  }
  // u -> vbuf (overwrites h1; same-wave DS ops are in-order)
#pragma unroll
  for (int nt = 0; nt < 2; ++nt) {
    int n0 = nt * 16;
#pragma unroll
    for (int g = 0; g < 8; ++g) {
      vbuf[(g + 8 * lhi) * HID + n0 + lrow] = t[nt][0][g];
      vbuf[(16 + g + 8 * lhi) * HID + n0 + lrow] = t[nt][1][g];
    }
  }

  // ---- dh1 = U @ W2^T ; v = dh1 * silu'(a1) ----
#pragma unroll
  for (int nt = 0; nt < 2; ++nt) {
    int n0 = nt * 16;
    v8f c0 = zero8(), c1 = zero8();
#pragma unroll
    for (int kk = 0; kk < HID; kk += 4) {
      v2f b = ldB_tr(W2, HID, kk, n0, lrow, lhi);
      c0 = wmma_f32(ldA_s(vbuf, 0, kk, lrow, lhi), b, c0);
      c1 = wmma_f32(ldA_s(vbuf, 16, kk, lrow, lhi), b, c1);
    }
#pragma unroll
    for (int g = 0; g < 8; ++g) {
      c0[g] *= dsiluf(a1[nt][0][g]);
      c1[g] *= dsiluf(a1[nt][1][g]);
    }
    t[nt][0] = c0;
    t[nt][1] = c1;
  }
  // v -> vbuf (overwrites u)
#pragma unroll
  for (int nt = 0; nt < 2; ++nt) {
    int n0 = nt * 16;
#pragma unroll
    for (int g = 0; g < 8; ++g) {
      vbuf[(g + 8 * lhi) * HID + n0 + lrow] = t[nt][0][g];
      vbuf[(16 + g + 8 * lhi) * HID + n0 + lrow] = t[nt][1][g];
    }
  }
}

__global__ __launch_bounds__(256) void score_grad_kernel(
    const float* __restrict__ z_, const float* __restrict__ gW1_,
    const float* __restrict__ gb1_, const float* __restrict__ gW2_,
    const float* __restrict__ gb2_, const float* __restrict__ gW3_,
    const float* __restrict__ W_, float* __restrict__ out_) {
  __shared__ float vbuf_all[8][MT * HID];  // 32 KB: per-wave activation scratch

  // Pin address spaces so the backend emits global_load/ds_load (not flat).
  gfp z   = (gfp)z_;
  gfp gW1 = (gfp)gW1_;
  gfp gb1 = (gfp)gb1_;
  gfp gW2 = (gfp)gW2_;
  gfp gb2 = (gfp)gb2_;
  gfp gW3 = (gfp)gW3_;
  gfp W   = (gfp)W_;
  gfpw out = (gfpw)out_;

  const int tid = threadIdx.x;
  const int wid = tid >> 5;       // wave id 0..7 (wave32)
  const int lane = tid & 31;
  const int lrow = lane & 15;
  const int lhi = lane >> 4;
  const int R = blockIdx.x * MT;  // this workgroup's batch-row base
  sfpw vbuf = (sfpw)&vbuf_all[wid][0];

  if (wid < 4) {
    // Waves 0..3: two full output blocks each (k = 2w, 2w+1):
    //   out_k = -( V_k @ W1_k^T  +  Z8 @ W[7-k] )
#pragma unroll
    for (int jj = 0; jj < 2; ++jj) {
      const int k = wid * 2 + jj;
      mlp_v(z, gW1, gb1, gW2, gb2, gW3, k, R, lrow, lhi, vbuf);
      gfp Wk = W + (size_t)(7 - k) * NN * NN;  // row-major [256][256]
      gfp Z8 = z + (size_t)PMAX * NN;
      gfp W1 = gW1 + (size_t)k * NN * HID;
      for (int nt = 0; nt < 16; ++nt) {
        const int n0 = nt * 16;
        v8f c0 = zero8(), c1 = zero8();
#pragma unroll 4
        for (int kk = 0; kk < NN; kk += 4) {
          v2f b = ldB_rm(Wk, NN, kk, n0, lrow, lhi);
          c0 = wmma_f32(ldA_g(Z8, ZW, R, kk, lrow, lhi), b, c0);
          c1 = wmma_f32(ldA_g(Z8, ZW, R + 16, kk, lrow, lhi), b, c1);
        }
#pragma unroll
        for (int kk = 0; kk < HID; kk += 4) {  // mlp grad: V @ W1^T
          v2f b = ldB_tr(W1, HID, kk, n0, lrow, lhi);
          c0 = wmma_f32(ldA_s(vbuf, 0, kk, lrow, lhi), b, c0);
          c1 = wmma_f32(ldA_s(vbuf, 16, kk, lrow, lhi), b, c1);
        }
#pragma unroll
        for (int g = 0; g < 8; ++g) {
          out[(size_t)(R + g + 8 * lhi) * ZW + k * NN + n0 + lrow] = -c0[g];
          out[(size_t)(R + 16 + g + 8 * lhi) * ZW + k * NN + n0 + lrow] = -c1[g];
        }
      }
    }
  } else {
    // Waves 4..7: block 8, one 64-column N-chunk each.
    //   out_8 = -( V_8 @ W1_8^T + sum_r Z_{7-r} @ W[r]^T )  (single GEMM, K=2048)
    const int nt0 = (wid - 4) * 4;
    mlp_v(z, gW1, gb1, gW2, gb2, gW3, PMAX, R, lrow, lhi, vbuf);
    gfp W1 = gW1 + (size_t)PMAX * NN * HID;
    for (int nt = nt0; nt < nt0 + 4; ++nt) {
      const int n0 = nt * 16;
      v8f c0 = zero8(), c1 = zero8();
#pragma unroll
      for (int r = 0; r < PMAX; ++r) {
        gfp Zl = z + (size_t)(7 - r) * NN;      // A rows: lag block 7-r
        gfp Wr = W + (size_t)r * NN * NN;       // B[k][n] = W[r][n][k]
#pragma unroll 4
        for (int j = 0; j < NN; j += 4) {
          v2f b = ldB_tr(Wr, NN, j, n0, lrow, lhi);  // contiguous pair in W[r] row
          c0 = wmma_f32(ldA_g(Zl, ZW, R, j, lrow, lhi), b, c0);
          c1 = wmma_f32(ldA_g(Zl, ZW, R + 16, j, lrow, lhi), b, c1);
        }
      }
#pragma unroll
      for (int kk = 0; kk < HID; kk += 4) {  // mlp grad for block 8
        v2f b = ldB_tr(W1, HID, kk, n0, lrow, lhi);
        c0 = wmma_f32(ldA_s(vbuf, 0, kk, lrow, lhi), b, c0);
        c1 = wmma_f32(ldA_s(vbuf, 16, kk, lrow, lhi), b, c1);
      }
#pragma unroll
      for (int g = 0; g < 8; ++g) {
        out[(size_t)(R + g + 8 * lhi) * ZW + PMAX * NN + n0 + lrow] = -c0[g];
        out[(size_t)(R + 16 + g + 8 * lhi) * ZW + PMAX * NN + n0 + lrow] = -c1[g];
      }
    }
  }
}

extern "C" void kernel_launch(void* const* d_in, const int* in_sizes, int n_in,
                              void* d_out, int out_size, void* d_ws, size_t ws_size,
                              hipStream_t stream) {
  const float* z   = (const float*)d_in[0];
  const float* gW1 = (const float*)d_in[1];
  const float* gb1 = (const float*)d_in[2];
  const float* gW2 = (const float*)d_in[3];
  const float* gb2 = (const float*)d_in[4];
  const float* gW3 = (const float*)d_in[5];
  // d_in[6] = gb3: constant offset, drops out of grad_z
  const float* W   = (const float*)d_in[7];
  float* out = (float*)d_out;

  score_grad_kernel<<<BATCH / MT, 256, 0, stream>>>(z, gW1, gb1, gW2, gb2, gW3, W, out);
}